// HF_28063316312693
// MI455X (gfx1250) — compile-verified
//
#include <hip/hip_runtime.h>

typedef float v4f __attribute__((ext_vector_type(4)));

#define HH_L          64   // row length (floats), fixed by the reference
#define ROWS_PER_PASS 16   // 256 threads / 16 lanes-per-row
#define ITERS         4    // passes per block -> 64 contiguous rows per block

// ds_swizzle_b32 with immediate BitMasked pattern: lane <- lane ^ xor_mask.
// Patterns (offset[14:10]=xor_mask, offset[4:0]=0x1F):
//   xor 1 -> 0x041F, xor 2 -> 0x081F, xor 4 -> 0x101F, xor 8 -> 0x201F
template <int PAT>
__device__ __forceinline__ float swz_add(float x) {
  return x + __int_as_float(
      __builtin_amdgcn_ds_swizzle(__float_as_int(x), PAT));
}

// One row is owned by 16 consecutive lanes, each holding one float4, so a
// wave32's b128 load covers 512 contiguous bytes (2 full rows). All traffic
// is non-temporal: 768 MiB streamed exactly once.
__global__ __launch_bounds__(256, 1) void hh_reflect_kernel(
    const float* __restrict__ v,
    const float* __restrict__ z,
    float* __restrict__ out,
    int nrows)
{
  const int quad = threadIdx.x & 15;  // which float4 of the row
  int row = blockIdx.x * (ROWS_PER_PASS * ITERS) + (threadIdx.x >> 4);

#pragma unroll
  for (int it = 0; it < ITERS; ++it, row += ROWS_PER_PASS) {
    if (row >= nrows) break;

    const long long base = (long long)row * HH_L + quad * 4;

    // Prefetch 2 passes (8 KB/stream) ahead of this block's cursor — far
    // enough to cover HBM latency, near enough to survive in cache.
    // Speculative OOB prefetch is silently dropped (ISA 10.5).
    // Emits global_prefetch_b8 (gfx1250 path).
    __builtin_prefetch(v + base + 2 * ROWS_PER_PASS * HH_L, 0, 0);
    __builtin_prefetch(z + base + 2 * ROWS_PER_PASS * HH_L, 0, 0);

    v4f v4 = __builtin_nontemporal_load((const v4f*)(v + base));
    v4f z4 = __builtin_nontemporal_load((const v4f*)(z + base));

    // Per-lane partial dots (4 elements each).
    float pvz = v4.x * z4.x + v4.y * z4.y + v4.z * z4.z + v4.w * z4.w;
    float pvv = v4.x * v4.x + v4.y * v4.y + v4.z * v4.z + v4.w * v4.w;

    // Butterfly reduction over the 16 lanes owning this row via immediate
    // ds_swizzle (no index VGPRs, no lane-bound guards). XOR masks <= 8
    // never flip lane bit 4, so the two rows in a wave32 stay independent.
    pvz = swz_add<0x041F>(pvz);
    pvz = swz_add<0x081F>(pvz);
    pvz = swz_add<0x101F>(pvz);
    pvz = swz_add<0x201F>(pvz);

    pvv = swz_add<0x041F>(pvv);
    pvv = swz_add<0x081F>(pvv);
    pvv = swz_add<0x101F>(pvv);
    pvv = swz_add<0x201F>(pvv);

    const float scale = 2.0f * pvz / pvv;  // 2 (v.z) / ||v||^2

    v4f o;
    o.x = fmaf(-scale, v4.x, z4.x);
    o.y = fmaf(-scale, v4.y, z4.y);
    o.z = fmaf(-scale, v4.z, z4.z);
    o.w = fmaf(-scale, v4.w, z4.w);

    __builtin_nontemporal_store(o, (v4f*)(out + base));
  }
}

extern "C" void kernel_launch(void* const* d_in, const int* in_sizes, int n_in,
                              void* d_out, int out_size, void* d_ws, size_t ws_size,
                              hipStream_t stream) {
  const float* v = (const float*)d_in[0];
  const float* z = (const float*)d_in[1];
  float* out     = (float*)d_out;

  const int nrows = in_sizes[0] / HH_L;  // 1,048,576 for the reference shape

  const int rowsPerBlock = ROWS_PER_PASS * ITERS;  // 64 contiguous rows/block
  int blocks = (nrows + rowsPerBlock - 1) / rowsPerBlock;  // 16384 blocks
  if (blocks < 1) blocks = 1;

  hh_reflect_kernel<<<blocks, 256, 0, stream>>>(v, z, out, nrows);
}